// ROICrop_25099788878505
// MI455X (gfx1250) — compile-verified
//
#include <hip/hip_runtime.h>
#include <hip/hip_bf16.h>
#include <stdint.h>

// ROI adaptive max-pool (ROIPool 7x7) for MI455X / gfx1250.
// Strategy: Tensor Data Mover stages each ROI crop (3-D tile: w x h x 4
// channels, row stride W, channel stride H*W) into LDS with a single
// tensor_load_to_lds; waves then do the 7x7 adaptive-max reduction from LDS.

#define POOL_OUT 7
#define NCH      4      // channels per block / per TDM tile
#define FM_B     2
#define FM_C     256
#define FM_H     50
#define FM_W     50

typedef unsigned int v4u __attribute__((ext_vector_type(4)));
typedef int          v8i __attribute__((ext_vector_type(8)));
typedef int          v4i __attribute__((ext_vector_type(4)));

__global__ __launch_bounds__(224)
void roipool_tdm_kernel(const float* __restrict__ fm,
                        const int*   __restrict__ rois,
                        float*       __restrict__ out) {
    __shared__ float tile[NCH * FM_H * FM_W];   // 40 KB worst case, packed w*h per channel

    const int r  = blockIdx.x;         // ROI index
    const int c0 = blockIdx.y * NCH;   // first channel of this block's chunk

    // ROI parameters — uniform across the block; force into SGPRs.
    const int b  = __builtin_amdgcn_readfirstlane(rois[r * 5 + 0]);
    const int x1 = __builtin_amdgcn_readfirstlane(rois[r * 5 + 1]);
    const int y1 = __builtin_amdgcn_readfirstlane(rois[r * 5 + 2]);
    const int x2 = __builtin_amdgcn_readfirstlane(rois[r * 5 + 3]);
    const int y2 = __builtin_amdgcn_readfirstlane(rois[r * 5 + 4]);
    const int w = x2 - x1;             // 1..50
    const int h = y2 - y1;             // 1..50

    if (threadIdx.x < 32) {
        // Wave 0 only: build the tensor DMA descriptor (D#) and fire the TDM.
        // Tile start address inside the feature map:
        const uint64_t gaddr = (uint64_t)(uintptr_t)
            (fm + (((uint64_t)(b * FM_C + c0)) * FM_H + (uint64_t)y1) * FM_W + (uint64_t)x1);
        const uint32_t laddr = (uint32_t)(uintptr_t)&tile[0];   // LDS byte offset

        // ---- D# group 0 (128b): count=1 | lds_addr | global_addr[56:0] | type=2
        v4u g0;
        g0[0] = 1u;                                   // count=1, user descriptor
        g0[1] = laddr;                                // lds_addr (bytes)
        g0[2] = (uint32_t)gaddr;                      // global_addr[31:0]
        g0[3] = (uint32_t)((gaddr >> 32) & 0x01FFFFFFu) | (2u << 30); // [56:32] | type=2

        // ---- D# group 1 (256b)
        v8i g1;
        g1[0] = (int)(2u << 16);                      // wg_mask=0 (not in cluster), data_size=4B
        g1[1] = (int)((uint32_t)w << 16);             // atomic_barrier=0 | tensor_dim0[15:0]=w
        g1[2] = (int)((uint32_t)h << 16);             // tensor_dim0[31:16]=0 | tensor_dim1[15:0]=h
        g1[3] = (int)((uint32_t)w << 16);             // tensor_dim1[31:16]=0 | tile_dim0=w
        g1[4] = (int)(((uint32_t)h) | ((uint32_t)NCH << 16)); // tile_dim1=h | tile_dim2=NCH
        g1[5] = (int)FM_W;                            // tensor_dim0_stride[31:0] = W (row stride)
        g1[6] = (int)(((uint32_t)(FM_H * FM_W) & 0xFFFFu) << 16); // d0s[47:32]=0 | d1s[15:0]
        g1[7] = (int)((uint32_t)(FM_H * FM_W) >> 16); // tensor_dim1_stride[47:16] (channel stride)

        // ---- D# group 2 (128b): tensor_dim2=NCH valid channels; tile_dim3=0 (unused)
        v4i g2;
        g2[0] = NCH; g2[1] = 0; g2[2] = 0; g2[3] = 0;
        // ---- D# group 3 (128b): unused dims
        v4i g3;
        g3[0] = 0; g3[1] = 0; g3[2] = 0; g3[3] = 0;
        // ---- 5th operand (int32x8): uncharacterized on this toolchain; zero-filled
        v8i g4;
        g4[0] = 0; g4[1] = 0; g4[2] = 0; g4[3] = 0;
        g4[4] = 0; g4[5] = 0; g4[6] = 0; g4[7] = 0;

        __builtin_amdgcn_tensor_load_to_lds(g0, g1, g2, g3, g4, /*cpol*/0);
        __builtin_amdgcn_s_wait_tensorcnt(0);
    }
    __syncthreads();   // LDS tile now valid for all 7 waves

    // One thread per (channel, oy, ox) bin: 4*49 = 196 bins.
    const int t = threadIdx.x;
    if (t < NCH * POOL_OUT * POOL_OUT) {
        const int cl  = t / (POOL_OUT * POOL_OUT);
        const int bin = t % (POOL_OUT * POOL_OUT);
        const int oy  = bin / POOL_OUT;
        const int ox  = bin % POOL_OUT;

        // torch adaptive_max_pool2d bin bounds (tile-local coordinates)
        const int xs = (ox * w) / POOL_OUT;
        const int xe = ((ox + 1) * w + POOL_OUT - 1) / POOL_OUT;
        const int ys = (oy * h) / POOL_OUT;
        const int ye = ((oy + 1) * h + POOL_OUT - 1) / POOL_OUT;

        const float* tc = &tile[cl * h * w];
        float m = -__builtin_inff();
        for (int y = ys; y < ye; ++y) {
            const float* row = tc + y * w;
            for (int x = xs; x < xe; ++x) {
                m = fmaxf(m, row[x]);
            }
        }
        out[(((size_t)r * FM_C + (size_t)(c0 + cl)) * POOL_OUT + oy) * POOL_OUT + ox] = m;
    }
}

extern "C" void kernel_launch(void* const* d_in, const int* in_sizes, int n_in,
                              void* d_out, int out_size, void* d_ws, size_t ws_size,
                              hipStream_t stream) {
    const float* fm   = (const float*)d_in[0];   // [2, 256, 50, 50] f32
    const int*   rois = (const int*)d_in[1];     // [R, 5] i32
    float*       out  = (float*)d_out;           // [R, 256, 7, 7] f32

    const int R = in_sizes[1] / 5;               // 192
    dim3 grid(R, FM_C / NCH);                    // 192 x 64 blocks
    dim3 block(224);                             // 7 wave32 waves
    roipool_tdm_kernel<<<grid, block, 0, stream>>>(fm, rois, out);
}